// MockLinearAttention_78408922955874
// MI455X (gfx1250) — compile-verified
//
#include <hip/hip_runtime.h>
#include <math.h>

typedef float v2f __attribute__((ext_vector_type(2)));
typedef float v8f __attribute__((ext_vector_type(8)));

#define HDIM 2048
#define SEQ  4096
#define BAT  8
#define NROWS (BAT * SEQ)      // 32768 token rows
#define PROJC 36               // 24 qkv + 8 z + 2 b + 2 a
#define WALLC 48               // padded to 3 WMMA N-tiles

__device__ __forceinline__ float silu_f(float x)    { return x / (1.f + __expf(-x)); }
__device__ __forceinline__ float sigmoid_f(float x) { return 1.f / (1.f + __expf(-x)); }
__device__ __forceinline__ float softplus_f(float x){ return (x > 20.f) ? x : log1pf(__expf(x)); }

// ---------------------------------------------------------------------------
// Kernel 0: pack w_qkv / w_z / w_b / w_a into zero-padded W_all[48][2048]
// ---------------------------------------------------------------------------
__global__ void build_wall(const float* __restrict__ wqkv, const float* __restrict__ wz,
                           const float* __restrict__ wb,   const float* __restrict__ wa,
                           float* __restrict__ wall) {
    int idx = blockIdx.x * 256 + threadIdx.x;
    if (idx >= WALLC * HDIM) return;
    int c = idx / HDIM, k = idx % HDIM;
    float v = 0.f;
    if      (c < 24) v = wqkv[c * HDIM + k];
    else if (c < 32) v = wz[(c - 24) * HDIM + k];
    else if (c < 34) v = wb[(c - 32) * HDIM + k];
    else if (c < 36) v = wa[(c - 34) * HDIM + k];
    wall[idx] = v;
}

// ---------------------------------------------------------------------------
// Kernel 1: proj[32768][36] = x[32768][2048] * W_all^T  via v_wmma_f32_16x16x4_f32
// One wave per 16-row M-tile; 3 N-tiles (48 cols) per wave; K loop in steps of 4.
// A frag (16x4 f32): lane L holds x[M = L&15][k + (L>>4)*2 + {0,1}]
// B frag (4x16 f32): lane L holds W_all[N = nt*16 + (L&15)][k + (L>>4)*2 + {0,1}]
// ---------------------------------------------------------------------------
__global__ void __launch_bounds__(32) proj_gemm(const float* __restrict__ x,
                                                const float* __restrict__ wall,
                                                float* __restrict__ proj) {
    const int lane  = threadIdx.x;
    const int m0    = blockIdx.x * 16;
    const int mrow  = lane & 15;
    const int khalf = (lane >> 4) * 2;          // 0 or 2

    const float* aptr = x    + (size_t)(m0 + mrow) * HDIM + khalf;
    const float* b0p  = wall + (size_t)( 0 + mrow) * HDIM + khalf;
    const float* b1p  = wall + (size_t)(16 + mrow) * HDIM + khalf;
    const float* b2p  = wall + (size_t)(32 + mrow) * HDIM + khalf;

    v8f d0 = {}, d1 = {}, d2 = {};
    #pragma unroll 4
    for (int k = 0; k < HDIM; k += 4) {
        v2f a  = *(const v2f*)(aptr + k);
        v2f w0 = *(const v2f*)(b0p + k);
        v2f w1 = *(const v2f*)(b1p + k);
        v2f w2 = *(const v2f*)(b2p + k);
        d0 = __builtin_amdgcn_wmma_f32_16x16x4_f32(false, a, false, w0, (short)0, d0, false, false);
        d1 = __builtin_amdgcn_wmma_f32_16x16x4_f32(false, a, false, w1, (short)0, d1, false, false);
        d2 = __builtin_amdgcn_wmma_f32_16x16x4_f32(false, a, false, w2, (short)0, d2, false, false);
    }

    // D frag: VGPR r -> row (r + 8*(lane>=16)), col = ntile*16 + (lane&15)
    const int rbase = (lane >> 4) * 8;
    #pragma unroll
    for (int r = 0; r < 8; ++r) {
        size_t row = (size_t)(m0 + r + rbase);
        proj[row * PROJC + ( 0 + mrow)] = d0[r];
        proj[row * PROJC + (16 + mrow)] = d1[r];
        if (mrow < 4)
            proj[row * PROJC + (32 + mrow)] = d2[r];
    }
}

// ---------------------------------------------------------------------------
// Kernel 2: per (batch, 16-row s-tile): conv+silu+gating+RMSNorm -> h[16][8],
// then out[16][2048] = h * w_out^T via 2 WMMAs per 16x16 N-tile (K=8).
// ---------------------------------------------------------------------------
__global__ void __launch_bounds__(256) fuse_out(const float* __restrict__ proj,
                                                const float* __restrict__ w_out,
                                                const float* __restrict__ conv_w,
                                                const float* __restrict__ dt_bias,
                                                const float* __restrict__ A_log,
                                                const float* __restrict__ norm_w,
                                                float* __restrict__ out) {
    __shared__ float sProj[19][PROJC];   // rows s0-3 .. s0+15
    __shared__ float sCore[16][9];       // stride 9: conflict-free
    __shared__ float sH[16][9];

    const int t   = threadIdx.x;
    const int tpb = SEQ / 16;                 // 256 tiles per batch
    const int b   = blockIdx.x / tpb;
    const int s0  = (blockIdx.x % tpb) * 16;

    // stage proj halo tile (zero left-pad at batch start)
    for (int i = t; i < 19 * PROJC; i += 256) {
        int j = i / PROJC, c = i % PROJC;
        int s = s0 - 3 + j;
        sProj[j][c] = (s >= 0) ? proj[((size_t)b * SEQ + s) * PROJC + c] : 0.f;
    }
    __syncthreads();

    // phase 1: causal conv (K=4) + silu, beta, g, core; stash gate*norm_w
    if (t < 128) {
        int srow = t >> 3, v = t & 7, n = v >> 2, d = v & 3;
        float m[3];
        #pragma unroll
        for (int i = 0; i < 3; ++i) {
            int c = i * 8 + v;                // q: c=v, k: 8+v, value: 16+v
            float acc = 0.f;
            #pragma unroll
            for (int kk = 0; kk < 4; ++kk)
                acc += conv_w[c * 4 + kk] * sProj[srow + kk][c];
            m[i] = silu_f(acc);
        }
        float z    = sProj[srow + 3][24 + v];
        float bb   = sProj[srow + 3][32 + n];
        float aa   = sProj[srow + 3][34 + n];
        float beta = sigmoid_f(bb);
        float g    = -__expf(A_log[n]) * softplus_f(aa + dt_bias[n]);
        sCore[srow][v] = m[2] + beta * (m[0] + m[1]) + g;
        sH[srow][v]    = silu_f(z) * norm_w[d];
    }
    __syncthreads();

    // phase 2: RMSNorm over DV=4 and apply gate
    if (t < 128) {
        int srow = t >> 3, v = t & 7, n = v >> 2;
        float c0 = sCore[srow][4 * n + 0];
        float c1 = sCore[srow][4 * n + 1];
        float c2 = sCore[srow][4 * n + 2];
        float c3 = sCore[srow][4 * n + 3];
        float var = 0.25f * (c0 * c0 + c1 * c1 + c2 * c2 + c3 * c3);
        sH[srow][v] = sCore[srow][v] * rsqrtf(var + 1e-6f) * sH[srow][v];
    }
    __syncthreads();

    // phase 3: out = h[16][8] * w_out^T, M=16, K=8, N=2048 (8 waves x 16 N-tiles)
    const int wv    = t >> 5;
    const int lane  = t & 31;
    const int mrow  = lane & 15;
    const int khalf = (lane >> 4) * 2;
    const int rbase = (lane >> 4) * 8;

    v2f a0 = { sH[mrow][khalf + 0],     sH[mrow][khalf + 1] };
    v2f a1 = { sH[mrow][4 + khalf + 0], sH[mrow][4 + khalf + 1] };

    for (int i = 0; i < 16; ++i) {
        int nt = wv * 16 + i;
        const float* bp = w_out + (size_t)(nt * 16 + mrow) * 8 + khalf;
        v2f w0 = *(const v2f*)(bp);
        v2f w1 = *(const v2f*)(bp + 4);
        v8f dfr = {};
        dfr = __builtin_amdgcn_wmma_f32_16x16x4_f32(false, a0, false, w0, (short)0, dfr, false, false);
        dfr = __builtin_amdgcn_wmma_f32_16x16x4_f32(false, a1, false, w1, (short)0, dfr, false, false);

        float* op = out + ((size_t)b * SEQ + s0 + rbase) * HDIM + nt * 16 + mrow;
        #pragma unroll
        for (int r = 0; r < 8; ++r)
            op[(size_t)r * HDIM] = dfr[r];
    }
}

// ---------------------------------------------------------------------------
extern "C" void kernel_launch(void* const* d_in, const int* in_sizes, int n_in,
                              void* d_out, int out_size, void* d_ws, size_t ws_size,
                              hipStream_t stream) {
    const float* x       = (const float*)d_in[0];
    const float* w_qkv   = (const float*)d_in[1];
    const float* w_z     = (const float*)d_in[2];
    const float* w_b     = (const float*)d_in[3];
    const float* w_a     = (const float*)d_in[4];
    const float* w_out   = (const float*)d_in[5];
    const float* conv_w  = (const float*)d_in[6];
    const float* dt_bias = (const float*)d_in[7];
    const float* A_log   = (const float*)d_in[8];
    const float* norm_w  = (const float*)d_in[9];
    float* out = (float*)d_out;

    float* wall = (float*)d_ws;                 // 48*2048 floats  (384 KB)
    float* proj = wall + WALLC * HDIM;          // 32768*36 floats (4.5 MB)

    build_wall<<<(WALLC * HDIM + 255) / 256, 256, 0, stream>>>(w_qkv, w_z, w_b, w_a, wall);
    proj_gemm<<<NROWS / 16, 32, 0, stream>>>(x, wall, proj);
    fuse_out<<<NROWS / 16, 256, 0, stream>>>(proj, w_out, conv_w, dt_bias, A_log, norm_w, out);
}